// GAT_77068893160139
// MI455X (gfx1250) — compile-verified
//
#include <hip/hip_runtime.h>
#include <hip/hip_bf16.h>

// ---------------------------------------------------------------------------
// 2-layer GAT for MI455X (gfx1250, wave32).
//   conv1: GEMM 50000x128 @ 128x128 via V_WMMA_F32_16X16X4_F32 (exact fp32),
//          then 3-pass atomic segment-softmax + scatter-aggregate.
//   conv2: tiny 16x40 GEMM per node (scalar), same 3-pass edge scheme,
//          accumulate into d_out, in-place log_softmax.
// ---------------------------------------------------------------------------

typedef float v2f __attribute__((ext_vector_type(2)));
typedef float v8f __attribute__((ext_vector_type(8)));

#define N_NODES 50000
#define N_EDGES 1000000
#define E_TOT   (N_EDGES + N_NODES)   // +self loops
#define F_IN    128
#define HID     16
#define HEADS   8
#define HH      (HEADS * HID)         // 128
#define N_CLS   40
#define NEG_SLOPE 0.2f
// enc(-inf): 0xFF800000 ^ 0x7FFFFFFF = 0x807FFFFF
#define NEG_INF_ENC (-2139095041)

__device__ __forceinline__ int fenc(float f) {
    int i = __float_as_int(f);
    return i >= 0 ? i : (i ^ 0x7FFFFFFF);
}
__device__ __forceinline__ float fdec(int k) {
    return __int_as_float(k >= 0 ? k : (k ^ 0x7FFFFFFF));
}

// ---------------- init ----------------
__global__ void k_fill_f32(float* __restrict__ p, float v, int n) {
    int i = blockIdx.x * blockDim.x + threadIdx.x;
    if (i < n) p[i] = v;
}
__global__ void k_fill_i32(int* __restrict__ p, int v, int n) {
    int i = blockIdx.x * blockDim.x + threadIdx.x;
    if (i < n) p[i] = v;
}

// ---------------- conv1 GEMM: H1[N,128] = X[N,128] @ W1[128,128] -----------
// wave w of 8 handles column tile [16w,16w+16); block handles 16 rows.
__global__ __launch_bounds__(256) void k_gemm1(const float* __restrict__ X,
                                               const float* __restrict__ W,
                                               float* __restrict__ H) {
    const int wave = threadIdx.x >> 5;
    const int lane = threadIdx.x & 31;
    const int half = lane >> 4;       // 0: K={0,1}, 1: K={2,3} within a k-step
    const int l15  = lane & 15;       // A: row M; B/D: col N
    const int m0   = blockIdx.x * 16;
    const int n0   = wave * 16;

    v8f c = {};
    for (int k = 0; k < F_IN; k += 4) {
        const int ka = k + 2 * half;
        v2f a, b;
        a.x = X[(m0 + l15) * F_IN + ka];
        a.y = X[(m0 + l15) * F_IN + ka + 1];
        b.x = W[(ka    ) * HH + n0 + l15];
        b.y = W[(ka + 1) * HH + n0 + l15];
        c = __builtin_amdgcn_wmma_f32_16x16x4_f32(
            /*neg_a=*/false, a, /*neg_b=*/false, b,
            /*c_mod=*/(short)0, c, /*reuse_a=*/false, /*reuse_b=*/false);
    }
    // D layout: VGPR i -> row i (lanes 0-15) / row i+8 (lanes 16-31), col = l15
    #pragma unroll
    for (int i = 0; i < 8; ++i) {
        const int row = i + half * 8;
        H[(m0 + row) * HH + n0 + l15] = c[i];
    }
}

// ---------------- conv1 attention dots: a_s/a_d [N, HEADS] ----------------
__global__ void k_attn1(const float* __restrict__ H,
                        const float* __restrict__ att_s,
                        const float* __restrict__ att_d,
                        float* __restrict__ as1, float* __restrict__ ad1) {
    int t = blockIdx.x * blockDim.x + threadIdx.x;
    if (t >= N_NODES * HEADS) return;
    int n = t / HEADS, h = t % HEADS;
    const float* hp = H + n * HH + h * HID;
    float s = 0.f, d = 0.f;
    #pragma unroll
    for (int c = 0; c < HID; ++c) {
        float v = hp[c];
        s += v * att_s[h * HID + c];
        d += v * att_d[h * HID + c];
    }
    as1[t] = s; ad1[t] = d;
}

__device__ __forceinline__ void edge_sd(const int* __restrict__ ei, int e, int& s, int& d) {
    if (e < N_EDGES) { s = ei[e]; d = ei[N_EDGES + e]; }
    else             { s = d = e - N_EDGES; }           // self loop
}

// ---------------- conv1 pass 1: segment max ----------------
__global__ void k_edge_max1(const int* __restrict__ ei,
                            const float* __restrict__ as1, const float* __restrict__ ad1,
                            int* __restrict__ m1) {
    int e = blockIdx.x * blockDim.x + threadIdx.x;
    if (e >= E_TOT) return;
    int s, d; edge_sd(ei, e, s, d);
    #pragma unroll
    for (int h = 0; h < HEADS; ++h) {
        float v = as1[s * HEADS + h] + ad1[d * HEADS + h];
        v = v > 0.f ? v : NEG_SLOPE * v;
        atomicMax(&m1[d * HEADS + h], fenc(v));
    }
}

// ---------------- conv1 pass 2: exp-sum denominators ----------------
__global__ void k_edge_sum1(const int* __restrict__ ei,
                            const float* __restrict__ as1, const float* __restrict__ ad1,
                            const int* __restrict__ m1, float* __restrict__ den1) {
    int e = blockIdx.x * blockDim.x + threadIdx.x;
    if (e >= E_TOT) return;
    int s, d; edge_sd(ei, e, s, d);
    #pragma unroll
    for (int h = 0; h < HEADS; ++h) {
        float v = as1[s * HEADS + h] + ad1[d * HEADS + h];
        v = v > 0.f ? v : NEG_SLOPE * v;
        float ex = expf(v - fdec(m1[d * HEADS + h]));
        atomicAdd(&den1[d * HEADS + h], ex);
    }
}

// ---------------- conv1 pass 3: scatter weighted messages ----------------
// one thread per (edge, head); 16 atomic adds each
__global__ void k_edge_acc1(const int* __restrict__ ei,
                            const float* __restrict__ as1, const float* __restrict__ ad1,
                            const int* __restrict__ m1, const float* __restrict__ den1,
                            const float* __restrict__ H, float* __restrict__ acc1) {
    long long t = (long long)blockIdx.x * blockDim.x + threadIdx.x;
    if (t >= (long long)E_TOT * HEADS) return;
    int e = (int)(t >> 3);
    int h = (int)(t & 7);
    int s, d; edge_sd(ei, e, s, d);
    float v = as1[s * HEADS + h] + ad1[d * HEADS + h];
    v = v > 0.f ? v : NEG_SLOPE * v;
    float ex = expf(v - fdec(m1[d * HEADS + h]));
    float alpha = ex / (den1[d * HEADS + h] + 1e-16f);
    const float* hp = H + s * HH + h * HID;
    float* op = acc1 + d * HH + h * HID;
    #pragma unroll
    for (int c = 0; c < HID; ++c)
        atomicAdd(&op[c], hp[c] * alpha);
}

// ---------------- conv1 epilogue: mean over heads + b1, ELU -> x2 ----------
__global__ void k_node_elu(const float* __restrict__ acc1, const float* __restrict__ b1,
                           float* __restrict__ x2) {
    int t = blockIdx.x * blockDim.x + threadIdx.x;
    if (t >= N_NODES * HID) return;
    int n = t / HID, c = t % HID;
    float s = 0.f;
    #pragma unroll
    for (int h = 0; h < HEADS; ++h) s += acc1[n * HH + h * HID + c];
    float v = s * (1.f / HEADS) + b1[c];
    x2[t] = v > 0.f ? v : (expf(v) - 1.f);
}

// ---------------- conv2 GEMM (16x40 per node) + attention dots ------------
__global__ void k_conv2_gemm(const float* __restrict__ x2, const float* __restrict__ W2,
                             const float* __restrict__ atts, const float* __restrict__ attd,
                             float* __restrict__ h2,
                             float* __restrict__ as2, float* __restrict__ ad2) {
    int n = blockIdx.x * blockDim.x + threadIdx.x;
    if (n >= N_NODES) return;
    float xr[HID];
    #pragma unroll
    for (int k = 0; k < HID; ++k) xr[k] = x2[n * HID + k];
    float s = 0.f, d = 0.f;
    for (int c = 0; c < N_CLS; ++c) {
        float acc = 0.f;
        #pragma unroll
        for (int k = 0; k < HID; ++k) acc += xr[k] * W2[k * N_CLS + c];
        h2[n * N_CLS + c] = acc;
        s += acc * atts[c];
        d += acc * attd[c];
    }
    as2[n] = s; ad2[n] = d;
}

// ---------------- conv2 edge passes (heads = 1) ----------------
__global__ void k_edge_max2(const int* __restrict__ ei,
                            const float* __restrict__ as2, const float* __restrict__ ad2,
                            int* __restrict__ m2) {
    int e = blockIdx.x * blockDim.x + threadIdx.x;
    if (e >= E_TOT) return;
    int s, d; edge_sd(ei, e, s, d);
    float v = as2[s] + ad2[d];
    v = v > 0.f ? v : NEG_SLOPE * v;
    atomicMax(&m2[d], fenc(v));
}

__global__ void k_edge_sum2(const int* __restrict__ ei,
                            const float* __restrict__ as2, const float* __restrict__ ad2,
                            const int* __restrict__ m2, float* __restrict__ den2) {
    int e = blockIdx.x * blockDim.x + threadIdx.x;
    if (e >= E_TOT) return;
    int s, d; edge_sd(ei, e, s, d);
    float v = as2[s] + ad2[d];
    v = v > 0.f ? v : NEG_SLOPE * v;
    atomicAdd(&den2[d], expf(v - fdec(m2[d])));
}

__global__ void k_edge_acc2(const int* __restrict__ ei,
                            const float* __restrict__ as2, const float* __restrict__ ad2,
                            const int* __restrict__ m2, const float* __restrict__ den2,
                            const float* __restrict__ h2, float* __restrict__ out) {
    int e = blockIdx.x * blockDim.x + threadIdx.x;
    if (e >= E_TOT) return;
    int s, d; edge_sd(ei, e, s, d);
    float v = as2[s] + ad2[d];
    v = v > 0.f ? v : NEG_SLOPE * v;
    float ex = expf(v - fdec(m2[d]));
    float alpha = ex / (den2[d] + 1e-16f);
    const float* hp = h2 + s * N_CLS;
    float* op = out + d * N_CLS;
    #pragma unroll
    for (int c = 0; c < N_CLS; ++c)
        atomicAdd(&op[c], hp[c] * alpha);
}

// ---------------- final: + b2, log_softmax (in place) ----------------
__global__ void k_final(float* __restrict__ out, const float* __restrict__ b2) {
    int n = blockIdx.x * blockDim.x + threadIdx.x;
    if (n >= N_NODES) return;
    float v[N_CLS];
    float mx = -3.4e38f;
    #pragma unroll
    for (int c = 0; c < N_CLS; ++c) {
        v[c] = out[n * N_CLS + c] + b2[c];
        mx = fmaxf(mx, v[c]);
    }
    float sum = 0.f;
    #pragma unroll
    for (int c = 0; c < N_CLS; ++c) sum += expf(v[c] - mx);
    float lse = logf(sum);
    #pragma unroll
    for (int c = 0; c < N_CLS; ++c) out[n * N_CLS + c] = v[c] - mx - lse;
}

// ---------------------------------------------------------------------------
extern "C" void kernel_launch(void* const* d_in, const int* in_sizes, int n_in,
                              void* d_out, int out_size, void* d_ws, size_t ws_size,
                              hipStream_t stream) {
    const float* x        = (const float*)d_in[0];
    const int*   ei       = (const int*)  d_in[1];
    const float* W1       = (const float*)d_in[2];
    const float* att_src1 = (const float*)d_in[3];
    const float* att_dst1 = (const float*)d_in[4];
    const float* b1       = (const float*)d_in[5];
    const float* W2       = (const float*)d_in[6];
    const float* att_src2 = (const float*)d_in[7];
    const float* att_dst2 = (const float*)d_in[8];
    const float* b2       = (const float*)d_in[9];
    float* out = (float*)d_out;

    // workspace layout (floats)
    float* ws   = (float*)d_ws;
    float* H1   = ws;                        // N*128
    float* as1  = H1  + N_NODES * HH;        // N*8
    float* ad1  = as1 + N_NODES * HEADS;     // N*8
    int*   m1   = (int*)(ad1 + N_NODES * HEADS);      // N*8
    float* den1 = (float*)(m1 + N_NODES * HEADS);     // N*8
    float* acc1 = den1 + N_NODES * HEADS;    // N*128
    float* x2   = acc1 + N_NODES * HH;       // N*16
    float* h2   = x2   + N_NODES * HID;      // N*40
    float* as2  = h2   + N_NODES * N_CLS;    // N
    float* ad2  = as2  + N_NODES;            // N
    int*   m2   = (int*)(ad2 + N_NODES);     // N
    float* den2 = (float*)(m2 + N_NODES);    // N

    const int B = 256;
    auto blk = [](long long n, int b) { return (unsigned)((n + b - 1) / b); };

    // init accumulators / maxima / output
    k_fill_f32<<<blk(N_NODES * HH, B),    B, 0, stream>>>(acc1, 0.f, N_NODES * HH);
    k_fill_f32<<<blk(N_NODES * HEADS, B), B, 0, stream>>>(den1, 0.f, N_NODES * HEADS);
    k_fill_i32<<<blk(N_NODES * HEADS, B), B, 0, stream>>>(m1, NEG_INF_ENC, N_NODES * HEADS);
    k_fill_f32<<<blk((long long)N_NODES * N_CLS, B), B, 0, stream>>>(out, 0.f, N_NODES * N_CLS);
    k_fill_f32<<<blk(N_NODES, B), B, 0, stream>>>(den2, 0.f, N_NODES);
    k_fill_i32<<<blk(N_NODES, B), B, 0, stream>>>(m2, NEG_INF_ENC, N_NODES);

    // conv1
    k_gemm1<<<N_NODES / 16, 256, 0, stream>>>(x, W1, H1);
    k_attn1<<<blk(N_NODES * HEADS, B), B, 0, stream>>>(H1, att_src1, att_dst1, as1, ad1);
    k_edge_max1<<<blk(E_TOT, B), B, 0, stream>>>(ei, as1, ad1, m1);
    k_edge_sum1<<<blk(E_TOT, B), B, 0, stream>>>(ei, as1, ad1, m1, den1);
    k_edge_acc1<<<blk((long long)E_TOT * HEADS, B), B, 0, stream>>>(ei, as1, ad1, m1, den1, H1, acc1);
    k_node_elu<<<blk(N_NODES * HID, B), B, 0, stream>>>(acc1, b1, x2);

    // conv2
    k_conv2_gemm<<<blk(N_NODES, B), B, 0, stream>>>(x2, W2, att_src2, att_dst2, h2, as2, ad2);
    k_edge_max2<<<blk(E_TOT, B), B, 0, stream>>>(ei, as2, ad2, m2);
    k_edge_sum2<<<blk(E_TOT, B), B, 0, stream>>>(ei, as2, ad2, m2, den2);
    k_edge_acc2<<<blk(E_TOT, B), B, 0, stream>>>(ei, as2, ad2, m2, den2, h2, out);

    // log_softmax
    k_final<<<blk(N_NODES, B), B, 0, stream>>>(out, b2);
}